// BaseTransform_77060303225117
// MI455X (gfx1250) — compile-verified
//
#include <hip/hip_runtime.h>
#include <hip/hip_bf16.h>

// ---------------------------------------------------------------------------
// BEV pool (LSS view transform) for MI455X / gfx1250.
//
//   x:  (1, 6, 118, 32, 88, 80) fp32   -> 1,993,728 points x 80 channels
//   out:(1, 80, 360, 360)       fp32   -> out[c*NV + flat], NV = 360*360
//
// Memory-bound: ~638 MB streamed + ~41.5 MB accumulated => ~30 us floor at
// 23.3 TB/s.  Three kernels:
//   0) zero the accumulator
//   1) geometry: fused per-camera 3x4 affine on (u'z, v'z, z, 1), batched
//      16 points/wave through V_WMMA_F32_16X16X4_F32; voxelize with
//      reciprocal multiplies (dx is compile-time constant) -> flat index.
//   2) scatter: per-wave double-buffered GLOBAL_LOAD_ASYNC_TO_LDS_B128
//      staging (ASYNCcnt-tracked, th:TH_LOAD_NT so the one-pass stream does
//      not evict the L2-resident accumulator), then 4 native
//      global_atomic_add_f32 per (point, channel-quad).
// ---------------------------------------------------------------------------

#define NCAM   6
#define DBINS  118
#define FH     32
#define FW     88
#define CCH    80
#define NX     360
#define NY     360
#define NZ     1
#define NVOX   (NZ * NX * NY)              // 129600
#define PPC    (DBINS * FH * FW)           // 332288 points per camera
#define NPTS   (NCAM * PPC)                // 1993728 points
#define OUTSZ  (CCH * NVOX)                // 10368000 floats
#define QUADS  (NPTS * 20u)                // 39,874,560 float4 work items
#define GROUPS (QUADS / 32u)               // 1,246,080 wave-groups (exact)
#define SCATTER_BLOCKS 2048u               // persistent blocks (8 waves each)

typedef float v2f  __attribute__((ext_vector_type(2)));
typedef float v4f  __attribute__((ext_vector_type(4)));
typedef float v8f  __attribute__((ext_vector_type(8)));

// ---------------------------------------------------------------------------
// Kernel 0: zero the output accumulator (harness poisons d_out).
// ---------------------------------------------------------------------------
__global__ __launch_bounds__(256) void bev_zero_kernel(v4f* __restrict__ out4)
{
    unsigned i = blockIdx.x * 256u + threadIdx.x;
    if (i < (OUTSZ / 4u)) {
        v4f z = {0.0f, 0.0f, 0.0f, 0.0f};
        out4[i] = z;
    }
}

// ---------------------------------------------------------------------------
// Kernel 1: geometry.  grid = (PPC/128, NCAM), block = 256 (8 waves).
// One V_WMMA_F32_16X16X4_F32 per wave transforms 16 points:
//   D(16x16) = A(16x4:[Mfull|tfull]) x B(4x16: (u'z, v'z, z, 1) per column)
// ---------------------------------------------------------------------------
__global__ __launch_bounds__(256) void bev_geom_kernel(
    const float* __restrict__ c2l,   // (1,N,4,4)
    const float* __restrict__ kin,   // (1,N,4,4)
    const float* __restrict__ img,   // (1,N,4,4)
    const float* __restrict__ lid,   // (1,4,4)
    int* __restrict__ vidx)          // (NPTS)
{
    const int n = blockIdx.y;                    // camera (uniform -> SGPR)
    const float* Mc = c2l + n * 16;
    const float* Mk = kin + n * 16;
    const float* Mi = img + n * 16;

    // --- uniform per-camera matrix algebra (scalarized by the compiler) ----
    const float r00 = Mc[0], r01 = Mc[1], r02 = Mc[2],  tr0 = Mc[3];
    const float r10 = Mc[4], r11 = Mc[5], r12 = Mc[6],  tr1 = Mc[7];
    const float r20 = Mc[8], r21 = Mc[9], r22 = Mc[10], tr2 = Mc[11];

    const float k00 = Mk[0], k01 = Mk[1], k02 = Mk[2];
    const float k10 = Mk[4], k11 = Mk[5], k12 = Mk[6];
    const float k20 = Mk[8], k21 = Mk[9], k22 = Mk[10];

    const float p00 = Mi[0], p01 = Mi[1], p02 = Mi[2],  pt0 = Mi[3];
    const float p10 = Mi[4], p11 = Mi[5], p12 = Mi[6],  pt1 = Mi[7];
    const float p20 = Mi[8], p21 = Mi[9], p22 = Mi[10], pt2 = Mi[11];

    const float e00 = lid[0], e01 = lid[1], e02 = lid[2],  et0 = lid[3];
    const float e10 = lid[4], e11 = lid[5], e12 = lid[6],  et1 = lid[7];
    const float e20 = lid[8], e21 = lid[9], e22 = lid[10], et2 = lid[11];

    // inv(intrinsics) via adjugate
    const float det = k00 * (k11 * k22 - k12 * k21)
                    - k01 * (k10 * k22 - k12 * k20)
                    + k02 * (k10 * k21 - k11 * k20);
    const float id = 1.0f / det;
    const float i00 = (k11 * k22 - k12 * k21) * id;
    const float i01 = (k02 * k21 - k01 * k22) * id;
    const float i02 = (k01 * k12 - k02 * k11) * id;
    const float i10 = (k12 * k20 - k10 * k22) * id;
    const float i11 = (k00 * k22 - k02 * k20) * id;
    const float i12 = (k02 * k10 - k00 * k12) * id;
    const float i20 = (k10 * k21 - k11 * k20) * id;
    const float i21 = (k01 * k20 - k00 * k21) * id;
    const float i22 = (k00 * k11 - k01 * k10) * id;

    // combine = rots @ inv(K)
    const float c00 = r00*i00 + r01*i10 + r02*i20;
    const float c01 = r00*i01 + r01*i11 + r02*i21;
    const float c02 = r00*i02 + r01*i12 + r02*i22;
    const float c10 = r10*i00 + r11*i10 + r12*i20;
    const float c11 = r10*i01 + r11*i11 + r12*i21;
    const float c12 = r10*i02 + r11*i12 + r12*i22;
    const float c20 = r20*i00 + r21*i10 + r22*i20;
    const float c21 = r20*i01 + r21*i11 + r22*i21;
    const float c22 = r20*i02 + r21*i12 + r22*i22;

    // Mfull = extra_rots @ combine ; tfull = extra_rots @ trans + extra_trans
    const float m00 = e00*c00 + e01*c10 + e02*c20;
    const float m01 = e00*c01 + e01*c11 + e02*c21;
    const float m02 = e00*c02 + e01*c12 + e02*c22;
    const float m10 = e10*c00 + e11*c10 + e12*c20;
    const float m11 = e10*c01 + e11*c11 + e12*c21;
    const float m12 = e10*c02 + e11*c12 + e12*c22;
    const float m20 = e20*c00 + e21*c10 + e22*c20;
    const float m21 = e20*c01 + e21*c11 + e22*c21;
    const float m22 = e20*c02 + e21*c12 + e22*c22;
    const float t0  = e00*tr0 + e01*tr1 + e02*tr2 + et0;
    const float t1  = e10*tr0 + e11*tr1 + e12*tr2 + et1;
    const float t2  = e20*tr0 + e21*tr1 + e22*tr2 + et2;

    // --- per-lane point prep ----------------------------------------------
    const int lane = threadIdx.x & 31;
    const int wave = threadIdx.x >> 5;
    const int row  = lane & 15;              // A-matrix row / point slot
    const bool hi  = lane >= 16;             // upper half-wave -> K=2,3 slots

    const int q  = (blockIdx.x * 8 + wave) * 16 + row;   // point id in camera
    const int w  = q % FW;
    const int hq = q / FW;
    const int h  = hq % FH;
    const int dd = hq / FH;

    const float u  = (float)w * (703.0f / 87.0f);   // linspace(0, IW-1, FW)
    const float v  = (float)h * (255.0f / 31.0f);   // linspace(0, IH-1, FH)
    const float zd = 1.0f + 0.5f * (float)dd;       // arange(1, 60, 0.5)

    // p1 = post_rots^T @ (frustum - post_trans)
    const float q0 = u  - pt0;
    const float q1 = v  - pt1;
    const float q2 = zd - pt2;
    const float p1x = p00*q0 + p10*q1 + p20*q2;
    const float p1y = p01*q0 + p11*q1 + p21*q2;
    const float p1z = p02*q0 + p12*q1 + p22*q2;

    const float pcx = p1x * p1z;   // homogeneous camera point (K=0)
    const float pcy = p1y * p1z;   // K=1
    const float pcz = p1z;         // K=2 ; K=3 = 1.0

    // A: 16x4, lanes 0-15 hold K=0(V0)/K=1(V1), lanes 16-31 hold K=2/K=3.
    const float a0r0 = hi ? m02 : m00;
    const float a0r1 = hi ? m12 : m10;
    const float a0r2 = hi ? m22 : m20;
    const float a1r0 = hi ? t0  : m01;
    const float a1r1 = hi ? t1  : m11;
    const float a1r2 = hi ? t2  : m21;
    v2f A;
    A.x = (row == 0) ? a0r0 : (row == 1) ? a0r1 : (row == 2) ? a0r2 : 0.0f;
    A.y = (row == 0) ? a1r0 : (row == 1) ? a1r1 : (row == 2) ? a1r2 : 0.0f;

    // B: 4x16, column = point, same K striping as A.
    v2f Bv;
    Bv.x = hi ? pcz : pcx;
    Bv.y = hi ? 1.0f : pcy;

    v8f acc = {0.0f, 0.0f, 0.0f, 0.0f, 0.0f, 0.0f, 0.0f, 0.0f};
    acc = __builtin_amdgcn_wmma_f32_16x16x4_f32(
        /*neg_a=*/false, A, /*neg_b=*/false, Bv,
        /*c_mod=*/(short)0, acc, /*reuse_a=*/false, /*reuse_b=*/false);

    if (!hi) {
        const float X = acc[0];   // D row 0, column = point = lane
        const float Y = acc[1];
        const float Z = acc[2];
        // dx is a compile-time constant: reciprocal multiply instead of the
        // 20-op IEEE v_div_scale/v_div_fixup sequence (sub-ulp difference
        // only at bin boundaries).  (int) truncates toward zero == astype.
        const int cx = (int)((X + 54.0f) * (1.0f / 0.3f));
        const int cy = (int)((Y + 54.0f) * (1.0f / 0.3f));
        const int cz = (int)((Z + 10.0f) * (1.0f / 20.0f));
        const bool kept = (cx >= 0) & (cx < NX) & (cy >= 0) & (cy < NY) &
                          (cz >= 0) & (cz < NZ);
        const int flat = (cz * NX + cx) * NY + cy;
        vidx[n * PPC + q] = kept ? flat : -1;
    }
}

// ---------------------------------------------------------------------------
// Kernel 2: scatter-add, persistent waves.  Each wave grid-strides over
// 32-quad groups (QUADS % 32 == 0, so EXEC stays full).  Each group's 512 B
// of features is staged LDS-side with GLOBAL_LOAD_ASYNC_TO_LDS_B128,
// double-buffered: batch i+1 is in flight while batch i feeds the atomics.
// Per-wave private staging => no barriers; async loads retire in order, so
// s_wait_asynccnt 1 after issuing i+1 guarantees batch i is resident.
// ---------------------------------------------------------------------------
__global__ __launch_bounds__(256) void bev_scatter_kernel(
    const float* __restrict__ x,     // (NPTS*80) fp32, viewed as quads
    const int* __restrict__ vidx,    // (NPTS)
    float* __restrict__ out)         // (CCH, NVOX)
{
    __shared__ v4f stage[8][2][32];          // 8 KB: 8 waves x 2 slots x 512 B
    const unsigned lane   = threadIdx.x & 31u;
    const unsigned wave   = threadIdx.x >> 5;
    const unsigned waveId = blockIdx.x * 8u + wave;
    const unsigned stride = SCATTER_BLOCKS * 8u;

    if (waveId >= GROUPS) return;

    const unsigned lds0 = (unsigned)(size_t)&stage[wave][0][lane];
    const unsigned lds1 = (unsigned)(size_t)&stage[wave][1][lane];

    // prime the pipeline with group 0
    {
        const unsigned off = (waveId * 32u + lane) * 16u;    // byte offset
        asm volatile("global_load_async_to_lds_b128 %0, %1, %2 th:TH_LOAD_NT"
                     :: "v"(lds0), "v"(off), "s"(x) : "memory");
    }

    unsigned g = waveId;
    unsigned slot = 0u;
    for (;;) {
        const unsigned gnext = g + stride;
        const bool more = gnext < GROUPS;
        if (more) {
            const unsigned off = (gnext * 32u + lane) * 16u;
            const unsigned ldsN = (slot == 0u) ? lds1 : lds0;
            asm volatile("global_load_async_to_lds_b128 %0, %1, %2 th:TH_LOAD_NT"
                         :: "v"(ldsN), "v"(off), "s"(x) : "memory");
            asm volatile("s_wait_asynccnt 0x1" ::: "memory");  // batch g done
        } else {
            asm volatile("s_wait_asynccnt 0x0" ::: "memory");
        }

        // consume group g from LDS
        const unsigned qd = g * 32u + lane;          // quad id
        const unsigned p  = qd / 20u;                // point id
        const unsigned c4 = qd - p * 20u;            // channel quad
        const int idx = vidx[p];
        const v4f f = stage[wave][slot][lane];
        if (idx >= 0) {
            float* base = out + (size_t)(c4 * 4u) * (size_t)NVOX + (size_t)idx;
            unsafeAtomicAdd(base,                    f.x);  // global_atomic_add_f32
            unsafeAtomicAdd(base + 1 * (size_t)NVOX, f.y);
            unsafeAtomicAdd(base + 2 * (size_t)NVOX, f.z);
            unsafeAtomicAdd(base + 3 * (size_t)NVOX, f.w);
        }
        if (!more) break;
        g = gnext;
        slot ^= 1u;
    }
}

// ---------------------------------------------------------------------------
extern "C" void kernel_launch(void* const* d_in, const int* in_sizes, int n_in,
                              void* d_out, int out_size, void* d_ws, size_t ws_size,
                              hipStream_t stream) {
    const float* x   = (const float*)d_in[0];
    const float* c2l = (const float*)d_in[1];
    const float* kin = (const float*)d_in[2];
    const float* img = (const float*)d_in[3];
    const float* lid = (const float*)d_in[4];
    float* out = (float*)d_out;
    int*   vidx = (int*)d_ws;                 // NPTS * 4 bytes ~ 8 MB

    // 1) zero accumulator
    {
        const unsigned n4 = OUTSZ / 4u;
        dim3 grid((n4 + 255u) / 256u);
        bev_zero_kernel<<<grid, 256, 0, stream>>>((v4f*)out);
    }
    // 2) geometry -> per-point voxel index
    {
        dim3 grid(PPC / 128, NCAM);           // 128 points per 256-thread block
        bev_geom_kernel<<<grid, 256, 0, stream>>>(c2l, kin, img, lid, vidx);
    }
    // 3) scatter-add features (persistent waves, async LDS staging)
    {
        dim3 grid(SCATTER_BLOCKS);
        bev_scatter_kernel<<<grid, 256, 0, stream>>>(x, vidx, out);
    }
    (void)in_sizes; (void)n_in; (void)out_size; (void)ws_size;
}